// AttentionMap_75505525063894
// MI455X (gfx1250) — compile-verified
//
#include <hip/hip_runtime.h>
#include <math.h>
#include <stdint.h>

// CDNA5 wave32 WMMA fragments for V_WMMA_F32_16X16X4_F32:
//   A (16x4 f32): 2 VGPRs/lane, B (4x16 f32): 2 VGPRs/lane, C/D (16x16 f32): 8 VGPRs/lane
typedef __attribute__((ext_vector_type(2))) float v2f;
typedef __attribute__((ext_vector_type(4))) float v4f;
typedef __attribute__((ext_vector_type(8))) float v8f;
typedef __attribute__((ext_vector_type(4))) int   v4i;

typedef __attribute__((address_space(1))) v4i gv4i;   // global v4i
typedef __attribute__((address_space(3))) v4i lv4i;   // LDS v4i

#define NB 2
#define NH 16
#define SQ 2048
#define DH 64
#define WAVES 4
#define QT 16                 // query rows per wave
#define BQ (WAVES*QT)         // 64 query rows per block
#define KT 16                 // key tile
#define NKT (SQ/KT)           // 128 key tiles
#define LK 68                 // padded f32 row stride for K/V LDS tiles (bank-conflict free)
#define LW 18                 // padded f32 row stride for per-wave weight tile

#if __has_builtin(__builtin_amdgcn_global_load_async_to_lds_b128) && \
    __has_builtin(__builtin_amdgcn_s_wait_asynccnt)
#define ASYNC_STAGE 1
#else
#define ASYNC_STAGE 0
#endif

#if ASYNC_STAGE
// 128 threads issue async copies for one 16x64 f32 tile (2x b128 per thread).
// ASYNCcnt += 2 per wave. No VGPR round-trip: memory -> LDS directly.
__device__ __forceinline__ void stage_async(const float* gsrc, float* ldst, int tid) {
  #pragma unroll
  for (int u = 0; u < 2; ++u) {
    int i = u * 128 + tid;
    int r = i >> 4, col = (i & 15) * 4;
    __builtin_amdgcn_global_load_async_to_lds_b128(
        (gv4i*)(uintptr_t)(gsrc + (size_t)r * DH + col),
        (lv4i*)(uintptr_t)(ldst + r * LK + col),
        0, 0);
  }
}
#else
__device__ __forceinline__ void stage_sync(const float* gsrc, float* ldst, int tid) {
  #pragma unroll
  for (int u = 0; u < 2; ++u) {
    int i = u * 128 + tid;
    int r = i >> 4, col = (i & 15) * 4;
    v4f d = *(const v4f*)(gsrc + (size_t)r * DH + col);
    *(v4f*)(ldst + r * LK + col) = d;   // LK%4==0 -> 16B aligned
  }
}
#endif

__global__ __launch_bounds__(128) void fused_attn(
    const float* __restrict__ Qg, const float* __restrict__ Kg,
    const float* __restrict__ Vg, const int* __restrict__ Mg,
    float* __restrict__ Og, float* __restrict__ Wg)
{
  __shared__ float ldsT[2][KT * LK];            // double-buffered K tile (pass1) / V tile (pass3)
  __shared__ float ldsW[WAVES][QT * LW];

  const int tid  = threadIdx.x;
  const int lane = tid & 31;
  const int wv   = tid >> 5;
  const int l16  = lane & 15;
  const int hi   = lane >> 4;     // 0: lanes 0-15 (rows j / K 0-1), 1: lanes 16-31 (rows j+8 / K 2-3)

  const int bh   = blockIdx.x / (SQ / BQ);
  const int qblk = blockIdx.x % (SQ / BQ);
  const int b    = bh / NH;
  const int q0   = qblk * BQ + wv * QT;   // first query row owned by this wave

  const float* Qb = Qg + (size_t)bh * SQ * DH;
  const float* Kb = Kg + (size_t)bh * SQ * DH;
  const float* Vb = Vg + (size_t)bh * SQ * DH;
  const int*   Mr = Mg + (size_t)b * SQ * SQ + (size_t)q0 * SQ;
  float*       Wr = Wg + (size_t)bh * SQ * SQ + (size_t)q0 * SQ;  // wave's 16x2048 slab
  float*       Or = Og + (size_t)bh * SQ * DH + (size_t)q0 * DH;

  // ---- preload all Q A-fragments (D=64 -> 16 depth-4 steps), kept in VGPRs ----
  v2f qa[DH / 4];
  {
    const float* qr = Qb + (size_t)(q0 + l16) * DH + hi * 2;
    #pragma unroll
    for (int ks = 0; ks < DH / 4; ++ks) {
      qa[ks].x = qr[ks * 4 + 0];
      qa[ks].y = qr[ks * 4 + 1];
    }
  }

  float rmax[8];
  #pragma unroll
  for (int j = 0; j < 8; ++j) rmax[j] = -3.0e38f;

  // ================= pass 1: raw masked scores -> weights region, row max =================
#if ASYNC_STAGE
  stage_async(Kb, ldsT[0], tid);                // prime tile 0
#endif
  for (int t = 0; t < NKT; ++t) {
#if ASYNC_STAGE
    if (t + 1 < NKT) {
      stage_async(Kb + (size_t)(t + 1) * KT * DH, ldsT[(t + 1) & 1], tid);
      __builtin_amdgcn_s_wait_asynccnt(2);      // tile t landed; t+1 still in flight
    } else {
      __builtin_amdgcn_s_wait_asynccnt(0);
    }
    const float* kt = ldsT[t & 1];
#else
    stage_sync(Kb + (size_t)t * KT * DH, ldsT[0], tid);
    const float* kt = ldsT[0];
#endif
    __syncthreads();

    v8f acc = {};
    #pragma unroll
    for (int ks = 0; ks < DH / 4; ++ks) {       // 16 WMMAs: S_tile += Q(16x4) * K^T(4x16)
      v2f bf;
      const float* kp = &kt[l16 * LK + ks * 4 + hi * 2];
      bf.x = kp[0];
      bf.y = kp[1];
      acc = __builtin_amdgcn_wmma_f32_16x16x4_f32(false, qa[ks], false, bf,
                                                  (short)0, acc, false, false);
    }

    const int colg = t * KT + l16;
    #pragma unroll
    for (int j = 0; j < 8; ++j) {               // C layout: VGPR j -> row j (+8 for hi lanes)
      int r = j + 8 * hi;
      int m = Mr[(size_t)r * SQ + colg];
      float s = acc[j] * 0.125f + ((m >= 1) ? 0.0f : -1.0e9f);
      rmax[j] = fmaxf(rmax[j], s);
      Wr[(size_t)r * SQ + colg] = s;            // raw scores as temp, re-read -> keep cached
    }
    __syncthreads();                            // buf[t&1] free for the t+2 issue
  }

  // row-max reduction across the 16 lanes of each half (masks 1..8 never cross halves)
  #pragma unroll
  for (int j = 0; j < 8; ++j) {
    float v = rmax[j];
    v = fmaxf(v, __shfl_xor(v, 1, 32));
    v = fmaxf(v, __shfl_xor(v, 2, 32));
    v = fmaxf(v, __shfl_xor(v, 4, 32));
    v = fmaxf(v, __shfl_xor(v, 8, 32));
    rmax[j] = v;
  }

  // ================= pass 2: row sums of exp(s - max) (L2-resident re-read) =================
  float rsum[8];
  #pragma unroll
  for (int j = 0; j < 8; ++j) rsum[j] = 0.0f;
  for (int t = 0; t < NKT; ++t) {
    const int colg = t * KT + l16;
    #pragma unroll
    for (int j = 0; j < 8; ++j) {
      int r = j + 8 * hi;
      float s = Wr[(size_t)r * SQ + colg];
      rsum[j] += __expf(s - rmax[j]);
    }
  }
  #pragma unroll
  for (int j = 0; j < 8; ++j) {
    float v = rsum[j];
    v += __shfl_xor(v, 1, 32);
    v += __shfl_xor(v, 2, 32);
    v += __shfl_xor(v, 4, 32);
    v += __shfl_xor(v, 8, 32);
    rsum[j] = 1.0f / v;
  }

  // ================= pass 3: final weights out + O = W * V via WMMA =================
  v8f oacc[4];
  #pragma unroll
  for (int n = 0; n < 4; ++n) { v8f z = {}; oacc[n] = z; }

  float* wl = ldsW[wv];
#if ASYNC_STAGE
  stage_async(Vb, ldsT[0], tid);                // prime V tile 0
#endif
  for (int t = 0; t < NKT; ++t) {
#if ASYNC_STAGE
    if (t + 1 < NKT)
      stage_async(Vb + (size_t)(t + 1) * KT * DH, ldsT[(t + 1) & 1], tid);
    const float* vt = ldsT[t & 1];
#else
    stage_sync(Vb + (size_t)t * KT * DH, ldsT[0], tid);
    const float* vt = ldsT[0];
#endif

    // normalized weights: the only HBM weights write (streaming -> non-temporal).
    // Overlaps the in-flight async V copy.
    const int colg = t * KT + l16;
    #pragma unroll
    for (int j = 0; j < 8; ++j) {
      int r = j + 8 * hi;
      float s = Wr[(size_t)r * SQ + colg];
      float w = __expf(s - rmax[j]) * rsum[j];
      __builtin_nontemporal_store(w, &Wr[(size_t)r * SQ + colg]);
      wl[r * LW + l16] = w;                     // C-layout -> LDS, re-read as A fragments
    }

#if ASYNC_STAGE
    if (t + 1 < NKT) __builtin_amdgcn_s_wait_asynccnt(2);
    else             __builtin_amdgcn_s_wait_asynccnt(0);
#endif
    __syncthreads();

    #pragma unroll
    for (int ks = 0; ks < 4; ++ks) {            // 16 WMMAs: O(16x64) += W(16x16) * V(16x64)
      v2f af;
      const float* ap = &wl[l16 * LW + ks * 4 + hi * 2];
      af.x = ap[0];
      af.y = ap[1];
      #pragma unroll
      for (int n = 0; n < 4; ++n) {
        v2f bf;
        bf.x = vt[(ks * 4 + hi * 2 + 0) * LK + n * 16 + l16];
        bf.y = vt[(ks * 4 + hi * 2 + 1) * LK + n * 16 + l16];
        oacc[n] = __builtin_amdgcn_wmma_f32_16x16x4_f32(false, af, false, bf,
                                                        (short)0, oacc[n], false, false);
      }
    }
    __syncthreads();                            // buf[t&1] + ldsW free for next iteration
  }

  #pragma unroll
  for (int n = 0; n < 4; ++n) {
    #pragma unroll
    for (int j = 0; j < 8; ++j) {
      int r = j + 8 * hi;
      __builtin_nontemporal_store(oacc[n][j], &Or[(size_t)r * DH + n * 16 + l16]);
    }
  }
}

extern "C" void kernel_launch(void* const* d_in, const int* in_sizes, int n_in,
                              void* d_out, int out_size, void* d_ws, size_t ws_size,
                              hipStream_t stream) {
  const float* Q = (const float*)d_in[0];
  const float* K = (const float*)d_in[1];
  const float* V = (const float*)d_in[2];
  const int*   M = (const int*)d_in[3];
  float* out = (float*)d_out;                          // [B,H,S,D] first
  float* wts = out + (size_t)NB * NH * SQ * DH;        // then [B,H,S,S]
  dim3 grid(NB * NH * (SQ / BQ));                      // 1024 blocks, 4 waves each
  fused_attn<<<grid, 128, 0, stream>>>(Q, K, V, M, out, wts);
}